// TTTProbe_31799937860113
// MI455X (gfx1250) — compile-verified
//
#include <hip/hip_runtime.h>
#include <hip/hip_bf16.h>
#include <math.h>
#include <stdint.h>

typedef __attribute__((ext_vector_type(2))) float v2f;
typedef __attribute__((ext_vector_type(8))) float v8f;

#define T_SEQ   8192
#define D_PHI   4096
#define D_H     64
#define D_MID   256
#define M_BLK   64
#define BK      32
#define LDS_STRIDE 36              // 32 + 4 pad: 144B row stride (16B aligned), no bank conflicts
#define LDS_TILE   (64 * LDS_STRIDE)   // floats per 64x32 tile (padded)

// ---------------------------------------------------------------------------
// CDNA5 async LDS-DMA: global -> LDS, tracked by ASYNCcnt (no VGPR staging).
// GV addressing: 64-bit global addr in VGPR pair, LDS byte address in VDST.
// ---------------------------------------------------------------------------
__device__ __forceinline__ void async_b128(uint32_t lds_byte, const float* g) {
    asm volatile("global_load_async_to_lds_b128 %0, %1, off"
                 :: "v"(lds_byte), "v"((unsigned long long)(uintptr_t)g)
                 : "memory");
}
__device__ __forceinline__ void wait_async_le12() {
    asm volatile("s_wait_asynccnt 0xc" ::: "memory");
}
__device__ __forceinline__ void wait_async_0() {
    asm volatile("s_wait_asynccnt 0x0" ::: "memory");
}

// ---------------------------------------------------------------------------
// Kernel 1: u_Q = phi @ thetaQ^T, u_K = phi @ thetaK^T via V_WMMA_F32_16X16X4_F32,
// double-buffered async global->LDS staging. Grid: T/64 blocks of 128 threads
// (4 waves). Wave w owns d_h columns [16w,16w+16) for BOTH Q and K.
// ---------------------------------------------------------------------------
__global__ __launch_bounds__(128) void ttt_proj_kernel(
    const float* __restrict__ phi,
    const float* __restrict__ thK,
    const float* __restrict__ thQ,
    float* __restrict__ uQ,
    float* __restrict__ uK)
{
    __shared__ __align__(16) float smem[2 * 3 * LDS_TILE];  // [buf][phi|thq|thk]

    const int tid   = threadIdx.x;
    const int lane  = tid & 31;
    const int wave  = tid >> 5;          // n-tile 0..3
    const int nb    = wave * 16;
    const int mlane = lane & 15;
    const int koff  = (lane >> 4) * 2;   // lanes 0-15 -> K=0,1 ; lanes 16-31 -> K=2,3
    const int m0    = blockIdx.x * M_BLK;

    // Per-thread DMA slice: row0 = tid>>3 (+16 per r), 16B column chunk c4.
    const int row0 = tid >> 3;
    const int c4   = (tid & 7) * 4;
    const uint32_t smem_base = (uint32_t)(uintptr_t)smem;   // LDS byte offset
    const uint32_t loff0     = (uint32_t)((row0 * LDS_STRIDE + c4) * 4);

    const float* phi_p = phi + (size_t)(m0 + row0) * D_PHI + c4;
    const float* thq_p = thQ + (size_t)row0 * D_PHI + c4;
    const float* thk_p = thK + (size_t)row0 * D_PHI + c4;

    auto issue_stage = [&](int buf, int k0) {
        const uint32_t base = smem_base + (uint32_t)(buf * 3 * LDS_TILE * 4);
#pragma unroll
        for (int r = 0; r < 4; ++r) {
            const uint32_t lo = base + loff0 + (uint32_t)(r * 16 * LDS_STRIDE * 4);
            const size_t   go = (size_t)r * 16 * D_PHI + k0;
            async_b128(lo,                    phi_p + go);   // phi tile
            async_b128(lo + LDS_TILE * 4,     thq_p + go);   // thetaQ tile
            async_b128(lo + 2 * LDS_TILE * 4, thk_p + go);   // thetaK tile
        }
    };

    v8f zero = {};
    v8f accQ[4], accK[4];
#pragma unroll
    for (int s = 0; s < 4; ++s) { accQ[s] = zero; accK[s] = zero; }

    constexpr int NSTAGES = D_PHI / BK;     // 128
    issue_stage(0, 0);                      // prologue: stage 0 -> buf 0

    for (int s = 0; s < NSTAGES; ++s) {
        __syncthreads();                    // all waves done reading buf (s+1)&1 (stage s-1 compute)
        if (s + 1 < NSTAGES) {
            issue_stage((s + 1) & 1, (s + 1) * BK);  // prefetch next stage
            wait_async_le12();              // in-order ASYNCcnt: stage s's 12 DMAs retired
        } else {
            wait_async_0();
        }
        __syncthreads();                    // every wave's slice of stage s is in LDS

        const float* phi_c = &smem[(s & 1) * 3 * LDS_TILE];
        const float* thq_c = phi_c + LDS_TILE;
        const float* thk_c = thq_c + LDS_TILE;

#pragma unroll
        for (int kk = 0; kk < BK; kk += 4) {
            // B frag (4x16, K x N): element(K=koff+v, N=mlane) = theta[nb+mlane][kk+koff+v]
            const v2f bq = *reinterpret_cast<const v2f*>(
                &thq_c[(nb + mlane) * LDS_STRIDE + kk + koff]);
            const v2f bk = *reinterpret_cast<const v2f*>(
                &thk_c[(nb + mlane) * LDS_STRIDE + kk + koff]);
#pragma unroll
            for (int t = 0; t < 4; ++t) {
                // A frag (16x4, M x K): row M=mlane, K=koff..koff+1
                const v2f a = *reinterpret_cast<const v2f*>(
                    &phi_c[(t * 16 + mlane) * LDS_STRIDE + kk + koff]);
                accQ[t] = __builtin_amdgcn_wmma_f32_16x16x4_f32(
                    false, a, false, bq, (short)0, accQ[t], false, false);
                accK[t] = __builtin_amdgcn_wmma_f32_16x16x4_f32(
                    false, a, false, bk, (short)0, accK[t], false, false);
            }
        }
    }

    // D layout: VGPR v -> (M = v + 8*(lane>>4), N = lane&15)
    const int mrow = (lane >> 4) * 8;
#pragma unroll
    for (int t = 0; t < 4; ++t) {
#pragma unroll
        for (int v = 0; v < 8; ++v) {
            int m = m0 + t * 16 + mrow + v;
            int n = nb + mlane;
            uQ[(size_t)m * D_H + n] = accQ[t][v];
            uK[(size_t)m * D_H + n] = accK[t][v];
        }
    }
}

// ---------------------------------------------------------------------------
// Kernel 2: sequential TTT scan. One block of 256 threads; thread j owns
// W1 column j (64 regs), b1[j], W2[j]. 3 barriers/step, shuffle reductions.
// ---------------------------------------------------------------------------
__device__ __forceinline__ float gelu_fwd_tanh(float x, float& t_out) {
    const float c0 = 0.79788456f;
    float t = tanhf(c0 * x * (1.0f + 0.044715f * x * x));
    t_out = t;
    return 0.5f * x * (1.0f + t);
}
__device__ __forceinline__ float gelu_bwd_from_t(float x, float t) {
    return 0.5f * x * ((1.0f - t * t) * (0.79788456f + 0.1070322243f * x * x))
         + 0.5f * (1.0f + t);
}
__device__ __forceinline__ float sigmoidf_(float x) {
    return 1.0f / (1.0f + __expf(-x));
}

__global__ __launch_bounds__(256) void ttt_scan_kernel(
    const float* __restrict__ uQ,
    const float* __restrict__ uK,
    const float* __restrict__ Cseq,
    const float* __restrict__ W10,
    const float* __restrict__ b10,
    const float* __restrict__ W20,
    const float* __restrict__ b20,
    const float* __restrict__ log_eta,
    float* __restrict__ out)
{
    __shared__ float uq_s[64];
    __shared__ float uk_s[64];
    __shared__ float redq[8];
    __shared__ float redk[8];

    const int j    = threadIdx.x;
    const int lane = j & 31;
    const int wave = j >> 5;

    float w1[64];
#pragma unroll
    for (int i = 0; i < 64; ++i) w1[i] = W10[i * D_MID + j];
    float b1 = b10[j];
    float w2 = W20[j];
    float b2 = b20[0];
    const float eta = __expf(log_eta[0]);
    float loss = 0.0f;

    for (int t = 0; t < T_SEQ; ++t) {
        __syncthreads();                       // prev-step readers done
        if (j < 64)        uq_s[j]       = uQ[t * 64 + j];
        else if (j < 128)  uk_s[j - 64]  = uK[t * 64 + (j - 64)];
        __syncthreads();

        float zq = b1, zk = b1;
#pragma unroll
        for (int i = 0; i < 64; ++i) {
            zq = fmaf(uq_s[i], w1[i], zq);
            zk = fmaf(uk_s[i], w1[i], zk);
        }
        float tq, tk;
        float gq = gelu_fwd_tanh(zq, tq);
        float x2 = gelu_fwd_tanh(zk, tk);
        float pq = gq * w2;
        float pk = x2 * w2;
#pragma unroll
        for (int off = 16; off > 0; off >>= 1) {   // wave32 reduce
            pq += __shfl_xor(pq, off, 32);
            pk += __shfl_xor(pk, off, 32);
        }
        if (lane == 0) { redq[wave] = pq; redk[wave] = pk; }
        __syncthreads();
        float sq = 0.0f, sk = 0.0f;
#pragma unroll
        for (int w = 0; w < 8; ++w) { sq += redq[w]; sk += redk[w]; }

        float s_val = sigmoidf_(sq + b2);
        float pred  = sigmoidf_(sk + b2);
        if (j == 0) {
            out[t] = s_val;
            float d = s_val - Cseq[t];
            loss = fmaf(d, d, loss);
        }
        float dZ2 = 2.0f * pred * pred * (1.0f - pred);
        float dZ1 = dZ2 * w2 * gelu_bwd_from_t(zk, tk);
        float ed1 = eta * dZ1;
#pragma unroll
        for (int i = 0; i < 64; ++i)
            w1[i] = fmaf(-ed1, uk_s[i], w1[i]);
        b1 -= ed1;
        w2 = fmaf(-eta * dZ2, x2, w2);
        b2 -= eta * dZ2;
    }
    if (j == 0) out[T_SEQ] = loss;
}

// ---------------------------------------------------------------------------
extern "C" void kernel_launch(void* const* d_in, const int* in_sizes, int n_in,
                              void* d_out, int out_size, void* d_ws, size_t ws_size,
                              hipStream_t stream) {
    const float* phi     = (const float*)d_in[0];
    const float* Cseq    = (const float*)d_in[1];
    const float* thK     = (const float*)d_in[2];
    const float* thQ     = (const float*)d_in[3];
    const float* W10     = (const float*)d_in[4];
    const float* b10     = (const float*)d_in[5];
    const float* W20     = (const float*)d_in[6];
    const float* b20     = (const float*)d_in[7];
    const float* log_eta = (const float*)d_in[8];
    float* out = (float*)d_out;

    float* uQ = (float*)d_ws;
    float* uK = uQ + (size_t)T_SEQ * D_H;

    ttt_proj_kernel<<<T_SEQ / M_BLK, 128, 0, stream>>>(phi, thK, thQ, uQ, uK);
    ttt_scan_kernel<<<1, 256, 0, stream>>>(uQ, uK, Cseq, W10, b10, W20, b20,
                                           log_eta, out);
}